// AFNO2D_channelfirst_19834158973499
// MI455X (gfx1250) — compile-verified
//
#include <hip/hip_runtime.h>
#include <hip/hip_bf16.h>

// AFNO2D on gfx1250: everything (FFT stages + block-diag complex MLP) expressed
// as fp32 GEMMs on v_wmma_f32_16x16x4_f32. Memory-bound problem (~1.1 GB moved,
// ~100 GFLOP => ~47us floor at 23.3 TB/s), so the fp32 matrix path preserves
// accuracy at zero cost. Input image staging uses CDNA5 async global->LDS DMA.

typedef float v2f __attribute__((ext_vector_type(2)));
typedef float v8f __attribute__((ext_vector_type(8)));

__device__ __forceinline__ v8f wmma4(v2f a, v2f b, v8f c) {
  // D = A(16x4) * B(4x16) + C(16x16), fp32
  return __builtin_amdgcn_wmma_f32_16x16x4_f32(false, a, false, b, (short)0, c,
                                               false, false);
}

constexpr int HH = 128, WW = 128, WFP = 80;      // Wf=65 padded to 80
constexpr int SPAT = WFP * HH;                   // 10240 spectrum points/channel
constexpr int IMG_P = 132;                       // LDS pitches (bank-conflict free)
constexpr int MID_P = 84;
constexpr int O1_P = 100;
constexpr int M2_P = 136;
constexpr float PI_OVER_64 = 0.049087385212340517f;   // 2*pi/128

// ---- twiddle tables in ws ----
// Fc[128][80], Fs[128][80]  : forward W rfft, scaled 1/128 (full ortho fold)
// Hc[128][128], Hs[128][128]: cos/sin(2pi h h'/128), signs applied at use
// Gc[80][128], Gs[80][128]  : inverse W (C2R with c_wf weights), scaled 1/128
__global__ void afno_gen_tables(float* __restrict__ tab) {
  const int i = blockIdx.x * 256 + threadIdx.x;          // 64 blocks -> 16384
  float* Fc = tab;
  float* Fs = tab + 10240;
  float* Hc = tab + 20480;
  float* Hs = tab + 36864;
  float* Gc = tab + 53248;
  float* Gs = tab + 63488;
  if (i < 16384) {
    const int h = i >> 7, hp = i & 127;
    const float th = PI_OVER_64 * (float)((h * hp) & 127);
    Hc[i] = cosf(th);
    Hs[i] = sinf(th);
  }
  if (i < 10240) {
    {   // Fc/Fs: row w (128), col wf (80)
      const int w = i / WFP, wf = i % WFP;
      if (wf <= 64) {
        const float th = PI_OVER_64 * (float)((w * wf) & 127);
        Fc[i] = cosf(th) * (1.0f / 128.0f);
        Fs[i] = -sinf(th) * (1.0f / 128.0f);
      } else { Fc[i] = 0.0f; Fs[i] = 0.0f; }
    }
    {   // Gc/Gs: row wf (80), col w (128); c_wf = 1 for DC/Nyquist else 2
      const int wf = i >> 7, w = i & 127;
      if (wf <= 64) {
        const float cw = (wf == 0 || wf == 64) ? 1.0f : 2.0f;
        const float th = PI_OVER_64 * (float)((wf * w) & 127);
        Gc[i] = cw * cosf(th) * (1.0f / 128.0f);
        Gs[i] = cw * sinf(th) * (1.0f / 128.0f);
      } else { Gc[i] = 0.0f; Gs[i] = 0.0f; }
    }
  }
}

// ---- forward rfft2 fused per (b,c) image; DFT-as-GEMM through LDS ----
__global__ __launch_bounds__(256) void afno_fwd_fft(
    const float* __restrict__ x, const float* __restrict__ tab,
    float* __restrict__ Sr, float* __restrict__ Si) {
  extern __shared__ float lds[];
  float* img  = lds;                       // [128][IMG_P]
  float* midr = lds + HH * IMG_P;          // [128][MID_P]  (h major, wf minor)
  float* midi = midr + HH * MID_P;
  const float* Fc = tab;
  const float* Fs = tab + 10240;
  const float* Hc = tab + 20480;
  const float* Hs = tab + 36864;

  const int img_id = blockIdx.x;           // b*768 + c
  const float* xim = x + (size_t)img_id * (HH * WW);
  const int tid = threadIdx.x;

  // Stage 0: async DMA image -> LDS, 16B/lane, no VGPR round trip (ASYNCcnt).
  // LDS row pitch 132 floats = 528 B (16B aligned), 4-float chunks per lane.
#pragma unroll
  for (int it = 0; it < 16; ++it) {
    const int i4 = it * 256 + tid;                 // 0..4095 float4 chunks
    const int h = i4 >> 5, w4 = (i4 & 31) << 2;
    const unsigned ldsOff =
        (unsigned)(unsigned long long)(img + h * IMG_P + w4);
    const float* gsrc = xim + h * WW + w4;
    asm volatile("global_load_async_to_lds_b128 %0, %1, off"
                 :: "v"(ldsOff), "v"(gsrc) : "memory");
  }
  asm volatile("s_wait_asynccnt 0x0" ::: "memory");
  __syncthreads();

  const int lane = tid & 31, wave = tid >> 5;
  const int lm = lane & 15;
  const int kkh = (lane >> 4) << 1;        // K offset 0/2 (A,B operand halves)
  const int du  = (lane >> 4) << 3;        // D row offset 0/8

  // Stage 1: mid[h][wf] = sum_w img[h][w] * F[w][wf]   (real input)
  for (int job = wave; job < 80; job += 8) {
    const int part = job >= 40;            // 0 real (Fc), 1 imag (Fs)
    const int t = part ? job - 40 : job;
    const int mb = (t / 5) * 16, nb = (t % 5) * 16;
    const float* Bm = part ? Fs : Fc;
    v8f acc = {};
    for (int kc = 0; kc < WW; kc += 4) {
      v2f a, b;
      a.x = img[(mb + lm) * IMG_P + kc + kkh];
      a.y = img[(mb + lm) * IMG_P + kc + kkh + 1];
      b.x = Bm[(kc + kkh) * WFP + nb + lm];
      b.y = Bm[(kc + kkh + 1) * WFP + nb + lm];
      acc = wmma4(a, b, acc);
    }
    float* dst = part ? midi : midr;
#pragma unroll
    for (int j = 0; j < 8; ++j)
      dst[(mb + du + j) * MID_P + nb + lm] = acc[j];
  }
  __syncthreads();

  // Stage 2: S[wf][h'] = sum_h mid[h][wf] * e^{-i 2pi h h'/128}
  const size_t ob = (size_t)img_id * SPAT;
  for (int job = wave; job < 80; job += 8) {
    const int part = job >= 40;            // 0 real, 1 imag
    const int t = part ? job - 40 : job;
    const int mb = (t / 8) * 16, nb = (t % 8) * 16;
    v8f acc = {};
    for (int kc = 0; kc < HH; kc += 4) {
      v2f ar, ai, b0, b1;
      ar.x = midr[(kc + kkh) * MID_P + mb + lm];
      ar.y = midr[(kc + kkh + 1) * MID_P + mb + lm];
      ai.x = midi[(kc + kkh) * MID_P + mb + lm];
      ai.y = midi[(kc + kkh + 1) * MID_P + mb + lm];
      const float hc0 = Hc[(kc + kkh) * HH + nb + lm];
      const float hc1 = Hc[(kc + kkh + 1) * HH + nb + lm];
      const float hs0 = Hs[(kc + kkh) * HH + nb + lm];
      const float hs1 = Hs[(kc + kkh + 1) * HH + nb + lm];
      if (!part) {  // Re: Ar*Hc + Ai*Hs
        b0.x = hc0;  b0.y = hc1;  acc = wmma4(ar, b0, acc);
        b1.x = hs0;  b1.y = hs1;  acc = wmma4(ai, b1, acc);
      } else {      // Im: Ai*Hc - Ar*Hs
        b0.x = hc0;  b0.y = hc1;  acc = wmma4(ai, b0, acc);
        b1.x = -hs0; b1.y = -hs1; acc = wmma4(ar, b1, acc);
      }
    }
    float* dst = part ? Si : Sr;
#pragma unroll
    for (int j = 0; j < 8; ++j)
      dst[ob + (size_t)(mb + du + j) * HH + nb + lm] = acc[j];
  }
}

// ---- block-diagonal complex MLP + gating, in-place on spectrum planes ----
__global__ __launch_bounds__(256) void afno_mix(
    const float* __restrict__ w1, const float* __restrict__ b1,
    const float* __restrict__ w2, const float* __restrict__ b2,
    float* __restrict__ Sr, float* __restrict__ Si) {
  extern __shared__ float lds[];
  const int lane = threadIdx.x & 31, wave = threadIdx.x >> 5;
  float* o1r = lds + wave * (2 * 16 * O1_P);   // per-wave private tile
  float* o1i = o1r + 16 * O1_P;
  const int lm = lane & 15;
  const int kkh = (lane >> 4) << 1;
  const int du  = (lane >> 4) << 3;

  const int job = blockIdx.x * 8 + wave;       // 20480 jobs = 4*8*640
  const int m0 = (job % 640) * 16;             // spatial tile
  const int bk = job / 640;                    // b*8 + k
  const int k  = bk & 7;
  const size_t plane = (size_t)bk * 96 * SPAT; // = (b*768 + k*96)*SPAT

  // A operand: X tile [M=16 spatial][K=96 channels], kept in registers
  v2f ar[24], ai[24];
#pragma unroll
  for (int kc = 0; kc < 24; ++kc) {
    const size_t a0 = plane + (size_t)(kc * 4 + kkh) * SPAT + m0 + lm;
    ar[kc].x = Sr[a0];  ar[kc].y = Sr[a0 + SPAT];
    ai[kc].x = Si[a0];  ai[kc].y = Si[a0 + SPAT];
  }

  const float* w1r = w1 + (size_t)k * 9216;
  const float* w1i = w1 + (size_t)(8 + k) * 9216;
  const float* w2r = w2 + (size_t)k * 9216;
  const float* w2i = w2 + (size_t)(8 + k) * 9216;

  // layer 1: o1 = leaky(X * W1 + b1)   (complex, K=96)
  for (int nc = 0; nc < 6; ++nc) {
    const int o = nc * 16 + lm;
    const float brv = b1[k * 96 + o];
    const float biv = b1[768 + k * 96 + o];
    v8f accr = {brv, brv, brv, brv, brv, brv, brv, brv};
    v8f acci = {biv, biv, biv, biv, biv, biv, biv, biv};
    for (int kc = 0; kc < 24; ++kc) {
      const int i0 = kc * 4 + kkh;
      v2f wr, wi;
      wr.x = w1r[i0 * 96 + o];  wr.y = w1r[(i0 + 1) * 96 + o];
      wi.x = w1i[i0 * 96 + o];  wi.y = w1i[(i0 + 1) * 96 + o];
      const v2f win = -wi;                       // f32 WMMA has no B-negate
      accr = wmma4(ar[kc], wr, accr);
      accr = wmma4(ai[kc], win, accr);
      acci = wmma4(ai[kc], wr, acci);
      acci = wmma4(ar[kc], wi, acci);
    }
#pragma unroll
    for (int j = 0; j < 8; ++j) {
      float r = accr[j];  r = (r >= 0.0f) ? r : 0.1f * r;
      float m = acci[j];  m = (m >= 0.0f) ? m : 0.1f * m;
      o1r[(du + j) * O1_P + o] = r;              // D-layout -> A-layout via LDS
      o1i[(du + j) * O1_P + o] = m;
    }
  }

  // layer 2 + softshrink + complex gate multiply (in place)
  for (int nc = 0; nc < 6; ++nc) {
    const int o = nc * 16 + lm;
    const float brv = b2[k * 96 + o];
    const float biv = b2[768 + k * 96 + o];
    v8f accr = {brv, brv, brv, brv, brv, brv, brv, brv};
    v8f acci = {biv, biv, biv, biv, biv, biv, biv, biv};
    for (int kc = 0; kc < 24; ++kc) {
      const int i0 = kc * 4 + kkh;
      v2f a2r, a2i, wr, wi;
      a2r.x = o1r[lm * O1_P + i0];  a2r.y = o1r[lm * O1_P + i0 + 1];
      a2i.x = o1i[lm * O1_P + i0];  a2i.y = o1i[lm * O1_P + i0 + 1];
      wr.x = w2r[i0 * 96 + o];  wr.y = w2r[(i0 + 1) * 96 + o];
      wi.x = w2i[i0 * 96 + o];  wi.y = w2i[(i0 + 1) * 96 + o];
      const v2f win = -wi;
      accr = wmma4(a2r, wr, accr);
      accr = wmma4(a2i, win, accr);
      acci = wmma4(a2i, wr, acci);
      acci = wmma4(a2r, wi, acci);
    }
    const size_t cb = plane + (size_t)o * SPAT + m0 + du;
#pragma unroll
    for (int j = 0; j < 8; ++j) {
      float gr = accr[j];
      gr = (gr > 0.01f) ? gr - 0.01f : ((gr < -0.01f) ? gr + 0.01f : 0.0f);
      float gi = acci[j];
      gi = (gi > 0.01f) ? gi - 0.01f : ((gi < -0.01f) ? gi + 0.01f : 0.0f);
      const float xr = Sr[cb + j];
      const float xi = Si[cb + j];
      Sr[cb + j] = gr * xr - gi * xi;
      Si[cb + j] = gr * xi + gi * xr;
    }
  }
}

// ---- inverse irfft2 fused per image + residual add ----
__global__ __launch_bounds__(256) void afno_inv_fft(
    const float* __restrict__ Sr, const float* __restrict__ Si,
    const float* __restrict__ tab, const float* __restrict__ x,
    float* __restrict__ out) {
  extern __shared__ float lds[];
  float* m2r = lds;                        // [80][M2_P]  (wf major, h minor)
  float* m2i = lds + WFP * M2_P;
  const float* Hc = tab + 20480;
  const float* Hs = tab + 36864;
  const float* Gc = tab + 53248;
  const float* Gs = tab + 63488;

  const int img_id = blockIdx.x;
  const size_t sb = (size_t)img_id * SPAT;
  const int tid = threadIdx.x;
  const int lane = tid & 31, wave = tid >> 5;
  const int lm = lane & 15;
  const int kkh = (lane >> 4) << 1;
  const int du  = (lane >> 4) << 3;

  // Stage 1: Z[wf][h] = sum_h' S[wf][h'] * e^{+i 2pi h h'/128}
  for (int job = wave; job < 80; job += 8) {
    const int part = job >= 40;
    const int t = part ? job - 40 : job;
    const int mb = (t / 8) * 16, nb = (t % 8) * 16;
    v8f acc = {};
    for (int kc = 0; kc < HH; kc += 4) {
      v2f ar, ai, b0, b1;
      const size_t r0 = sb + (size_t)(mb + lm) * HH + kc + kkh;
      ar.x = Sr[r0];  ar.y = Sr[r0 + 1];
      ai.x = Si[r0];  ai.y = Si[r0 + 1];
      const float hc0 = Hc[(kc + kkh) * HH + nb + lm];
      const float hc1 = Hc[(kc + kkh + 1) * HH + nb + lm];
      const float hs0 = Hs[(kc + kkh) * HH + nb + lm];
      const float hs1 = Hs[(kc + kkh + 1) * HH + nb + lm];
      if (!part) {  // Re: Ar*Hc - Ai*Hs
        b0.x = hc0;  b0.y = hc1;  acc = wmma4(ar, b0, acc);
        b1.x = -hs0; b1.y = -hs1; acc = wmma4(ai, b1, acc);
      } else {      // Im: Ai*Hc + Ar*Hs
        b0.x = hc0;  b0.y = hc1;  acc = wmma4(ai, b0, acc);
        b1.x = hs0;  b1.y = hs1;  acc = wmma4(ar, b1, acc);
      }
    }
    float* dst = part ? m2i : m2r;
#pragma unroll
    for (int j = 0; j < 8; ++j)
      dst[(mb + du + j) * M2_P + nb + lm] = acc[j];
  }
  __syncthreads();

  // Stage 2: out[h][w] = sum_wf Zr*Gc - Zi*Gs  (+ residual x)
  const size_t xb = (size_t)img_id * (HH * WW);
  for (int job = wave; job < 64; job += 8) {
    const int mb = (job / 8) * 16, nb = (job % 8) * 16;
    v8f acc = {};
    for (int kc = 0; kc < WFP; kc += 4) {
      v2f ar, ai, b0, b1;
      ar.x = m2r[(kc + kkh) * M2_P + mb + lm];
      ar.y = m2r[(kc + kkh + 1) * M2_P + mb + lm];
      ai.x = m2i[(kc + kkh) * M2_P + mb + lm];
      ai.y = m2i[(kc + kkh + 1) * M2_P + mb + lm];
      b0.x = Gc[(kc + kkh) * WW + nb + lm];
      b0.y = Gc[(kc + kkh + 1) * WW + nb + lm];
      b1.x = -Gs[(kc + kkh) * WW + nb + lm];
      b1.y = -Gs[(kc + kkh + 1) * WW + nb + lm];
      acc = wmma4(ar, b0, acc);
      acc = wmma4(ai, b1, acc);
    }
#pragma unroll
    for (int j = 0; j < 8; ++j) {
      const size_t idx = xb + (size_t)(mb + du + j) * WW + nb + lm;
      out[idx] = acc[j] + x[idx];
    }
  }
}

extern "C" void kernel_launch(void* const* d_in, const int* in_sizes, int n_in,
                              void* d_out, int out_size, void* d_ws, size_t ws_size,
                              hipStream_t stream) {
  const float* x  = (const float*)d_in[0];   // [4,768,128,128]
  const float* w1 = (const float*)d_in[1];   // [2,8,96,96]
  const float* b1 = (const float*)d_in[2];   // [2,8,96]
  const float* w2 = (const float*)d_in[3];   // [2,8,96,96]
  const float* b2 = (const float*)d_in[4];   // [2,8,96]
  float* out = (float*)d_out;                // [4,768,128,128]

  // ws layout: [0,1MB) twiddle tables; then Sr, Si spectrum planes
  // (each 3072*80*128 floats = 125.8 MB; total required ~= 253 MB)
  float* tab = (float*)d_ws;
  float* Sr  = (float*)((char*)d_ws + (1u << 20));
  float* Si  = Sr + (size_t)3072 * SPAT;

  afno_gen_tables<<<64, 256, 0, stream>>>(tab);
  afno_fwd_fft<<<3072, 256, (size_t)(HH * IMG_P + 2 * HH * MID_P) * sizeof(float),
                stream>>>(x, tab, Sr, Si);
  afno_mix<<<2560, 256, (size_t)(8 * 2 * 16 * O1_P) * sizeof(float),
            stream>>>(w1, b1, w2, b2, Sr, Si);
  afno_inv_fft<<<3072, 256, (size_t)(2 * WFP * M2_P) * sizeof(float),
                stream>>>(Sr, Si, tab, x, out);
}